// VolumetricRenderer_82085414961261
// MI455X (gfx1250) — compile-verified
//
#include <hip/hip_runtime.h>

// CDNA5 / gfx1250 volumetric renderer.
// One ray per lane; per-ray affine coefficients computed with
// V_WMMA_F32_16X16X4_F32 ([N,6]x[6,8] GEMM); 128-bin loop uses native
// exp2/log2/tanh transcendentals (6 trans ops per bin).

typedef float v2f __attribute__((ext_vector_type(2)));
typedef float v8f __attribute__((ext_vector_type(8)));

#define TN_F 2.0f
#define TF_F 6.0f
#define FAR_DELTA_F 1e10f
#define L2E_F 1.44269504088896340736f

__device__ __forceinline__ float tanh_fast(float x) {
#if __has_builtin(__builtin_amdgcn_tanhf)
  return __builtin_amdgcn_tanhf(x);
#elif __has_builtin(__builtin_amdgcn_tanh_f32)
  return __builtin_amdgcn_tanh_f32(x);
#else
  float r;
  // v_nop covers the 1-op TRANS result hazard (ISA 7.4 case 3).
  asm("v_tanh_f32 %0, %1\n\tv_nop" : "=v"(r) : "v"(x));
  return r;
#endif
}

__launch_bounds__(256)
__global__ void volrender_kernel(const float* __restrict__ ray_o,
                                 const float* __restrict__ ray_d,
                                 const float* __restrict__ w_d,
                                 const float* __restrict__ b_d,
                                 const float* __restrict__ w_c,
                                 const float* __restrict__ b_c,
                                 const int* __restrict__ n_bins_p,
                                 float* __restrict__ out,
                                 int N)
{
  __shared__ float s_ray[256][6];   // staged (ox,oy,oz,dx,dy,dz) per ray
  __shared__ float s_coef[256][8];  // WMMA output: 8 affine coefs per ray
  __shared__ float s_w[12];         // w_d[0..2], w_c[0..8] (row-major [k][j])

  const int tid  = threadIdx.x;
  const int lane = tid & 31;
  const int wave = tid >> 5;
  const int half = lane >> 4;   // 0: lanes 0-15, 1: lanes 16-31
  const int l16  = lane & 15;
  const int ray  = blockIdx.x * 256 + tid;
  const bool valid = ray < N;

  if (tid < 3) s_w[tid]     = w_d[tid];
  if (tid < 9) s_w[3 + tid] = w_c[tid];

  if (valid) {
    s_ray[tid][0] = ray_o[ray * 3 + 0];
    s_ray[tid][1] = ray_o[ray * 3 + 1];
    s_ray[tid][2] = ray_o[ray * 3 + 2];
    s_ray[tid][3] = ray_d[ray * 3 + 0];
    s_ray[tid][4] = ray_d[ray * 3 + 1];
    s_ray[tid][5] = ray_d[ray * 3 + 2];
  } else {
    #pragma unroll
    for (int c = 0; c < 6; ++c) s_ray[tid][c] = 0.0f;
  }
  __syncthreads();

  // ---- Build B operands (4x16, K x N). Layout: VGPR v holds row K=v for
  // lanes 0-15 and row K=v+2 for lanes 16-31 (mirrors the documented C/D
  // split). Column map: n=0 -> a0 (o.w_d), n=1 -> a1 (d.w_d),
  // n=2..4 -> c0_j (o.w_c[:,j]), n=5..7 -> c1_j (d.w_c[:,j]).
  v2f B1, B2;
  {
    const int n = l16;
    #pragma unroll
    for (int v = 0; v < 2; ++v) {
      const int K = v + 2 * half;
      float e1 = 0.0f, e2 = 0.0f;
      if (K < 3) {
        if (n == 0)                 e1 = s_w[K];
        else if (n >= 2 && n <= 4)  e1 = s_w[3 + K * 3 + (n - 2)];
        if (n == 1)                 e2 = s_w[K];
        else if (n >= 5 && n <= 7)  e2 = s_w[3 + K * 3 + (n - 5)];
      }
      if (v == 0) { B1.x = e1; B2.x = e2; } else { B1.y = e1; B2.y = e2; }
    }
  }

  // ---- Two 16-ray tiles per wave; two chained WMMAs each:
  // D = A_o*B1 + A_d*B2  ->  8 coefficients per ray (columns 0..7).
  #pragma unroll
  for (int tile = 0; tile < 2; ++tile) {
    const int rowBase = wave * 32 + tile * 16;
    const int m = rowBase + l16;
    // A layout (16x4 f32): lanes 0-15 hold K=0,1 in VGPR0,1; lanes 16-31 K=2,3.
    v2f Ao, Ad;
    if (half == 0) {
      Ao.x = s_ray[m][0]; Ao.y = s_ray[m][1];   // K0=ox, K1=oy
      Ad.x = s_ray[m][3]; Ad.y = s_ray[m][4];
    } else {
      Ao.x = s_ray[m][2]; Ao.y = 0.0f;          // K2=oz, K3=0 (pad)
      Ad.x = s_ray[m][5]; Ad.y = 0.0f;
    }
    v8f D = {0.0f, 0.0f, 0.0f, 0.0f, 0.0f, 0.0f, 0.0f, 0.0f};
    D = __builtin_amdgcn_wmma_f32_16x16x4_f32(false, Ao, false, B1,
                                              (short)0, D, false, false);
    D = __builtin_amdgcn_wmma_f32_16x16x4_f32(false, Ad, false, B2,
                                              (short)0, D, false, false);
    // D layout: lane L, VGPR r -> (M = r + 8*(L/16), N = L%16). Keep N<8.
    if (l16 < 8) {
      #pragma unroll
      for (int r = 0; r < 8; ++r)
        s_coef[rowBase + 8 * half + r][l16] = D[r];
    }
  }
  __syncthreads();

  const float A0  = s_coef[tid][0] + b_d[0];   // density base
  const float A1  = s_coef[tid][1];            // density slope (per t)
  const float C00 = s_coef[tid][2] + b_c[0];
  const float C01 = s_coef[tid][3] + b_c[1];
  const float C02 = s_coef[tid][4] + b_c[2];
  const float C10 = s_coef[tid][5];
  const float C11 = s_coef[tid][6];
  const float C12 = s_coef[tid][7];

  const int   nb = *n_bins_p;
  const float dt = (TF_F - TN_F) / (float)(nb - 1);

  float t = TN_F;
  float trans = 1.0f;               // running exclusive transmittance
  float o0 = 0.0f, o1 = 0.0f, o2 = 0.0f;

  #pragma unroll 4
  for (int b = 0; b < nb; ++b) {
    const float delta = (b == nb - 1) ? FAR_DELTA_F : dt;
    // T = exp(-delta*softplus(x)) = 2^(-delta * log2(1 + 2^(x*log2e)))
    const float x = fmaf(t, A1, A0);
    const float u = __builtin_amdgcn_exp2f(x * L2E_F);
    const float g = __builtin_amdgcn_logf(1.0f + u);     // log2(1+e^x)
    const float T = __builtin_amdgcn_exp2f(-delta * g);
    const float wgt = trans * (1.0f - T);
    trans *= T;
    // sigmoid(y) = 0.5 + 0.5*tanh(y/2): one v_tanh_f32 per channel
    const float c0 = fmaf(tanh_fast(fmaf(t, C10, C00) * 0.5f), 0.5f, 0.5f);
    const float c1 = fmaf(tanh_fast(fmaf(t, C11, C01) * 0.5f), 0.5f, 0.5f);
    const float c2 = fmaf(tanh_fast(fmaf(t, C12, C02) * 0.5f), 0.5f, 0.5f);
    o0 = fmaf(wgt, c0, o0);
    o1 = fmaf(wgt, c1, o1);
    o2 = fmaf(wgt, c2, o2);
    t += dt;
  }

  if (valid) {
    out[ray * 3 + 0] = o0;
    out[ray * 3 + 1] = o1;
    out[ray * 3 + 2] = o2;
  }
}

extern "C" void kernel_launch(void* const* d_in, const int* in_sizes, int n_in,
                              void* d_out, int out_size, void* d_ws, size_t ws_size,
                              hipStream_t stream) {
  const float* ray_o = (const float*)d_in[0];
  const float* ray_d = (const float*)d_in[1];
  const float* w_d   = (const float*)d_in[2];
  const float* b_d   = (const float*)d_in[3];
  const float* w_c   = (const float*)d_in[4];
  const float* b_c   = (const float*)d_in[5];
  const int*   nb    = (const int*)d_in[6];
  float* out = (float*)d_out;

  const int N = in_sizes[0] / 3;          // ray_o is [N,3]
  const int blocks = (N + 255) / 256;     // 256 thr = 8 wave32 per block
  volrender_kernel<<<blocks, 256, 0, stream>>>(ray_o, ray_d, w_d, b_d,
                                               w_c, b_c, nb, out, N);
}